// GaussianBlurLayer_59450937312003
// MI455X (gfx1250) — compile-verified
//
#include <hip/hip_runtime.h>

// ---------------------------------------------------------------------------
// 5x5 separable Gaussian blur (sigma = 1.1, BORDER_REFLECT_101) on fp32 NHWC
// (64, 384, 512, 3).  Memory-bound: 302 MB min traffic -> ~13 us @ 23.3 TB/s.
// Vertical pass is expressed as banded matmuls on V_WMMA_F32_16X16X4_F32
// (D = Gv(16x20) * Xpad(20x16), 5 chained K=4 WMMAs per 16x16 tile);
// horizontal pass (stride-3 taps due to channel interleave) is VALU from LDS.
// ---------------------------------------------------------------------------

typedef __attribute__((ext_vector_type(2))) float v2f;
typedef __attribute__((ext_vector_type(8))) float v8f;

constexpr int BATCH = 64;
constexpr int IMH   = 384;
constexpr int IMW   = 512;
constexpr int CH    = 3;
constexpr int WC    = IMW * CH;        // 1536 flattened width*channels
constexpr int TILE_H = 16;             // WMMA M
constexpr int COLS_PER_WAVE = 64;      // output cols (flattened wc) per wave
constexpr int HALO = 8;                // >= 6 (2 px * 3 ch), rounded to 16-tiles
constexpr int LDS_COLS = COLS_PER_WAVE + 2 * HALO;  // 80 = 5 WMMA col tiles
constexpr int LDS_PITCH = LDS_COLS + 4;             // 84, de-conflict banks
constexpr int WAVES = 4;               // waves per block (128 threads)

__device__ __forceinline__ int refl101(int i, int n) {
  // reflect-101; halo <= 2 pixels so one reflection suffices
  i = (i < 0) ? -i : i;
  i = (i >= n) ? (2 * n - 2 - i) : i;
  return i;
}

__global__ __launch_bounds__(WAVES * 32)
void gauss5_wmma_f32(const float* __restrict__ x, float* __restrict__ out) {
  __shared__ float smem[WAVES * TILE_H * LDS_PITCH];

  const int lane = threadIdx.x & 31;
  const int wave = threadIdx.x >> 5;

  // Gaussian taps, sigma = 0.3*((5-1)*0.5 - 1) + 0.8 = 1.1  (2*sigma^2 = 2.42)
  const float e1 = __expf(-1.0f / 2.42f);
  const float e2 = __expf(-4.0f / 2.42f);
  const float s  = 1.0f + 2.0f * (e1 + e2);
  float wv[5];
  wv[0] = e2 / s; wv[1] = e1 / s; wv[2] = 1.0f / s; wv[3] = wv[1]; wv[4] = wv[0];

  // one wave-tile = 16 rows x 64 output cols; grid sized exactly, no guard
  const int colGroups = WC / COLS_PER_WAVE;    // 24
  const int rowTiles  = IMH / TILE_H;          // 24
  const int gwave = blockIdx.x * WAVES + wave;
  const int colg  = gwave % colGroups;
  const int tmp   = gwave / colGroups;
  const int rowt  = tmp % rowTiles;
  const int img   = tmp / rowTiles;

  const size_t imgbase = (size_t)img * IMH * WC;
  const int r0     = rowt * TILE_H - 2;              // padded-row origin
  const int wcLds0 = colg * COLS_PER_WAVE - HALO;    // LDS col origin (flattened)
  float* lds = &smem[wave * TILE_H * LDS_PITCH];

  const int n = lane & 15;   // N index (B/C/D column) for this lane
  const int h = lane >> 4;   // half-wave select
  const int m = n;           // A-matrix row (M) for this lane

  // ---------------- vertical pass: 5 WMMA column tiles of 16 ----------------
  for (int t = 0; t < 5; ++t) {
    // flattened column this lane owns in tile t; pixel-reflect along W keeping
    // the channel (wc = w*3 + ch); only non-identity at image edges.
    int wc = wcLds0 + 16 * t + n;
    int c3 = (wc + 12) % 3;                  // channel (wc >= -8 always)
    int px = (wc - c3) / 3;                  // exact multiple -> trunc ok
    px = refl101(px, IMW);
    const float* colp = x + imgbase + ((size_t)px * CH + c3);

    v8f acc = {0.f, 0.f, 0.f, 0.f, 0.f, 0.f, 0.f, 0.f};
#pragma unroll
    for (int c = 0; c < 5; ++c) {
      // A fragment: banded Gaussian Gv[i][p] = wv[p-i], p in [i, i+4].
      // 32-bit A 16x4 layout (ISA 7.12.2): lane=M, VGPR0 holds K={0,2},
      // VGPR1 holds K={1,3} split across lane halves.
      int d0 = 4 * c + 2 * h - m;
      int d1 = d0 + 1;
      v2f a;
      a.x = (d0 >= 0 && d0 < 5) ? wv[d0] : 0.0f;
      a.y = (d1 >= 0 && d1 < 5) ? wv[d1] : 0.0f;

      // B fragment: padded input rows (K) x 16 cols (N); assumed layout
      // mirrors A: VGPR0 -> K={0,2}, VGPR1 -> K={1,3} across lane halves.
      int rA = refl101(r0 + 4 * c + 2 * h,     IMH);
      int rB = refl101(r0 + 4 * c + 2 * h + 1, IMH);
      v2f b;
      b.x = colp[(size_t)rA * WC];
      b.y = colp[(size_t)rB * WC];

      acc = __builtin_amdgcn_wmma_f32_16x16x4_f32(
          /*neg_a=*/false, a, /*neg_b=*/false, b,
          /*c_mod=*/(short)0, acc, /*reuse_a=*/false, /*reuse_b=*/false);
    }

    // D layout (ISA 7.12.2): VGPR r -> row M = r + 8*h, col N = n
#pragma unroll
    for (int r = 0; r < 8; ++r) {
      lds[(r + 8 * h) * LDS_PITCH + 16 * t + n] = acc[r];
    }
  }

  __syncthreads();   // orders own ds stores vs. loads (waves are independent)

  // ---------------- horizontal pass: taps at stride 3 (NHWC) ----------------
  const int outRow0 = rowt * TILE_H;
  const int outCol0 = colg * COLS_PER_WAVE;
#pragma unroll 4
  for (int it = 0; it < (TILE_H * COLS_PER_WAVE) / 32; ++it) {
    int idx = lane + 32 * it;          // lanes cover 32 consecutive -> coalesced
    int i = idx >> 6;                  // row in tile
    int j = idx & 63;                  // col in tile
    // lds col for tap k: (j + 3*(k-2)) + HALO = j + 2 + 3k
    const float* row = &lds[i * LDS_PITCH + j + 2];
    float v = wv[0] * row[0] + wv[1] * row[3] + wv[2] * row[6] +
              wv[3] * row[9] + wv[4] * row[12];
    out[imgbase + (size_t)(outRow0 + i) * WC + (outCol0 + j)] = v;
  }
}

extern "C" void kernel_launch(void* const* d_in, const int* in_sizes, int n_in,
                              void* d_out, int out_size, void* d_ws, size_t ws_size,
                              hipStream_t stream) {
  (void)in_sizes; (void)n_in; (void)out_size; (void)d_ws; (void)ws_size;
  const float* x = (const float*)d_in[0];
  float* out = (float*)d_out;

  const int totalWaves = BATCH * (IMH / TILE_H) * (WC / COLS_PER_WAVE); // 36864
  const int blocks = totalWaves / WAVES;                                // 9216
  gauss5_wmma_f32<<<dim3(blocks), dim3(WAVES * 32), 0, stream>>>(x, out);
}